// WrappedCBS_2637109920034
// MI455X (gfx1250) — compile-verified
//
#include <hip/hip_runtime.h>
#include <math.h>

// ---------------------------------------------------------------------------
// CBS (Convergent Born Series) solver, 2048x2048 complex64, 12 stages.
// FFT-2048 = Stockham radix-16 (WMMA) x2 + radix-8 (VALU), in LDS.
// Workgroup = 256 threads = 8 wave32s; each wave performs 16 DFT-16s as a
// complex 16x16x16 matmul via 16 chained V_WMMA_F32_16X16X4_F32.
// Global<->LDS tile movement uses the Tensor Data Mover (TENSOR_LOAD_TO_LDS /
// TENSOR_STORE_FROM_LDS, TENSORcnt) when available.
// Working set (u + t + sos + src ~ 100MB) fits in the 192MB L2.
// ---------------------------------------------------------------------------

typedef __attribute__((ext_vector_type(2))) float v2f;
typedef __attribute__((ext_vector_type(8))) float v8f;
typedef __attribute__((ext_vector_type(4))) unsigned int u32x4;
typedef __attribute__((ext_vector_type(8))) int i32x8;
typedef __attribute__((ext_vector_type(4))) int i32x4;

#define PI_F 3.14159265358979323846f

constexpr int   NN    = 2048;
constexpr float K0F   = 0.7905694150420949f;
constexpr float K0SQF = K0F * K0F;
constexpr int   NSTG  = 12;
constexpr int   TPB   = 256;    // 8 wave32s

// 2*pi*fftfreq(NN) for index i
__device__ __forceinline__ float kfreq(int i) {
    int s = (i < NN / 2) ? i : (i - NN);
    return 2.0f * PI_F * (float)s / (float)NN;
}

// ---------------------------------------------------------------------------
// Tensor Data Mover: 2D tile descriptor (data_size = 8B complex64 elements).
// tile = tile_d0 x tile_d1 elements, row stride in memory = srd elements.
// ---------------------------------------------------------------------------
#if __has_builtin(__builtin_amdgcn_tensor_load_to_lds)
#define HAVE_TDM 1
#if __has_include(<hip/amd_detail/amd_gfx1250_TDM.h>)
#define TDM_ARGS6 1
#endif
#endif

#if HAVE_TDM
__device__ __forceinline__ void tdm_tile(const void* gbase, unsigned lds_off,
                                         unsigned tile_d0, unsigned tile_d1,
                                         unsigned tens_d0, unsigned tens_d1,
                                         unsigned srd, bool store) {
    unsigned long long ga = (unsigned long long)(uintptr_t)gbase;
    u32x4 g0 = { 1u,                                    // count = 1 valid D#
                 lds_off,                               // LDS byte address
                 (unsigned)(ga & 0xffffffffu),          // global_addr[31:0]
                 (unsigned)((ga >> 32) & 0x1ffffffu) |  // global_addr[56:32]
                     (2u << 30) };                      // type = 2 ("image")
    i32x8 g1 = { (int)(3u << 16),                       // data_size = 8 bytes
                 (int)((tens_d0 & 0xffffu) << 16),      // tensor_dim0[15:0]
                 (int)(((tens_d0 >> 16) & 0xffffu) |    // tensor_dim0[31:16]
                       ((tens_d1 & 0xffffu) << 16)),    // tensor_dim1[15:0]
                 (int)(((tens_d1 >> 16) & 0xffffu) |    // tensor_dim1[31:16]
                       (tile_d0 << 16)),                // tile_dim0
                 (int)(tile_d1 & 0xffffu),              // tile_dim1 (tile_dim2=0)
                 (int)srd,                              // tensor_dim0_stride[31:0]
                 0, 0 };                                // stride hi / dim1_stride
    i32x4 z4 = { 0, 0, 0, 0 };
#if TDM_ARGS6
    i32x8 z8 = { 0, 0, 0, 0, 0, 0, 0, 0 };
    if (!store) __builtin_amdgcn_tensor_load_to_lds(g0, g1, z4, z4, z8, 0);
    else        __builtin_amdgcn_tensor_store_from_lds(g0, g1, z4, z4, z8, 0);
#else
    if (!store) __builtin_amdgcn_tensor_load_to_lds(g0, g1, z4, z4, 0);
    else        __builtin_amdgcn_tensor_store_from_lds(g0, g1, z4, z4, 0);
#endif
}
#endif // HAVE_TDM

__device__ __forceinline__ unsigned lds_byte_off(const void* p) {
    return (unsigned)(uintptr_t)p;   // low 32 bits of flat LDS addr = LDS offset
}

// ---- DFT-16 matrix fragments in the ISA 16x4 A-matrix layout ---------------
// A VGPR0: lanes 0-15 hold (m=lane, k=4c+0), lanes 16-31 hold (m=lane-16, k=4c+2)
// A VGPR1: same with k+1.  W[m][k] = exp(-2*pi*i*m*k/16).
struct DftFrag { v2f ar[4]; v2f ai[4]; v2f nai[4]; };

__device__ __forceinline__ void make_dft16_frags(int lane, DftFrag& f) {
    int mrow = lane & 15;
    int kb   = (lane >> 4) * 2;
#pragma unroll
    for (int ch = 0; ch < 4; ++ch) {
        int   k0 = 4 * ch + kb;
        float a0 = -2.0f * PI_F * (float)(mrow * k0) / 16.0f;
        float a1 = -2.0f * PI_F * (float)(mrow * (k0 + 1)) / 16.0f;
        float s0, c0, s1, c1;
        __sincosf(a0, &s0, &c0);
        __sincosf(a1, &s1, &c1);
        f.ar[ch]  = { c0,  c1 };
        f.ai[ch]  = { s0,  s1 };
        f.nai[ch] = { -s0, -s1 };
    }
}

__device__ __forceinline__ v8f wmma_acc(v2f a, v2f b, v8f c) {
    return __builtin_amdgcn_wmma_f32_16x16x4_f32(false, a, false, b, (short)0,
                                                 c, false, false);
}

// ---- One Stockham radix-16 stage via WMMA ----------------------------------
// q = wave*16 + column; gather a[q + t*(NN/16)], twiddle exp(-2pi*i*k*t/(16M)),
// DFT-16 of 16 columns == complex 16x16x16 matmul = 4 K-chunks x 4 WMMA.
// M=1 stage has k=0 -> unit twiddles (compile-time elided).
template <int M, int SA, int SB>
__device__ __forceinline__ void radix16_stage(const float2* __restrict__ a,
                                              float2* __restrict__ b,
                                              int tid, const DftFrag& f) {
    const int lane = tid & 31;
    const int wave = tid >> 5;
    const int col  = lane & 15;
    const int th   = (lane >> 4) * 2;     // t offset inside a K-chunk
    const int q    = wave * 16 + col;
    const int k    = q & (M - 1);
    const int j    = q / M;

    v8f accR = {};
    v8f accI = {};
#pragma unroll
    for (int ch = 0; ch < 4; ++ch) {
        int t0 = 4 * ch + th;
        int t1 = t0 + 1;
        float2 x0 = a[(q + t0 * (NN / 16)) * SA];
        float2 x1 = a[(q + t1 * (NN / 16)) * SA];
        float d0r, d0i, d1r, d1i;
        if (M == 1) {                     // unit twiddles on first stage
            d0r = x0.x; d0i = x0.y; d1r = x1.x; d1i = x1.y;
        } else {
            const float tw = -2.0f * PI_F / (16.0f * (float)M);
            float s0, c0, s1, c1;
            __sincosf(tw * (float)(k * t0), &s0, &c0);
            __sincosf(tw * (float)(k * t1), &s1, &c1);
            d0r = x0.x * c0 - x0.y * s0; d0i = x0.x * s0 + x0.y * c0;
            d1r = x1.x * c1 - x1.y * s1; d1i = x1.x * s1 + x1.y * c1;
        }
        v2f br = { d0r, d1r };
        v2f bi = { d0i, d1i };
        // Re += Wr*Br - Wi*Bi ; Im += Wi*Br + Wr*Bi
        accR = wmma_acc(f.ar[ch],  br, accR);
        accR = wmma_acc(f.nai[ch], bi, accR);
        accI = wmma_acc(f.ai[ch],  br, accI);
        accI = wmma_acc(f.ar[ch],  bi, accI);
    }
    // D layout: lanes 0-15 rows 0..7 of their column, lanes 16-31 rows 8..15
    const int rb   = (lane >> 4) * 8;
    const int base = j * (16 * M) + k;
#pragma unroll
    for (int jj = 0; jj < 8; ++jj) {
        int trow = rb + jj;
        b[(base + trow * M) * SB] = make_float2(accR[jj], accI[jj]);
    }
}

// ---- Radix-8 tail stage (m = 256), one DFT-8 per thread --------------------
template <int SA, int SB>
__device__ __forceinline__ void radix8_stage(const float2* __restrict__ a,
                                             float2* __restrict__ b, int tid) {
    float xr[8], xi[8];
#pragma unroll
    for (int t = 0; t < 8; ++t) {
        float2 v = a[(tid + t * 256) * SA];
        float s, c;
        __sincosf(-2.0f * PI_F * (float)(tid * t) / 2048.0f, &s, &c);
        xr[t] = v.x * c - v.y * s;
        xi[t] = v.x * s + v.y * c;
    }
    const float cr[8] = { 1.f, 0.70710678f, 0.f, -0.70710678f, -1.f, -0.70710678f, 0.f, 0.70710678f };
    const float ci[8] = { 0.f, -0.70710678f, -1.f, -0.70710678f, 0.f, 0.70710678f, 1.f, 0.70710678f };
#pragma unroll
    for (int o = 0; o < 8; ++o) {
        float yr = 0.f, yi = 0.f;
#pragma unroll
        for (int t = 0; t < 8; ++t) {
            int idx = (o * t) & 7;
            yr += xr[t] * cr[idx] - xi[t] * ci[idx];
            yi += xr[t] * ci[idx] + xi[t] * cr[idx];
        }
        b[(o * 256 + tid) * SB] = make_float2(yr, yi);
    }
}

// ---- Forward FFT of 2048 complex in LDS; ping-pong, result lands in `out` --
template <int SIN, int SOUT>
__device__ __forceinline__ void fft2048(float2* in, float2* out, int tid,
                                        const DftFrag& f) {
    __syncthreads();
    radix16_stage<1, SIN, SOUT>(in, out, tid, f);
    __syncthreads();
    radix16_stage<16, SOUT, SIN>(out, in, tid, f);
    __syncthreads();
    radix8_stage<SIN, SOUT>(in, out, tid);
    __syncthreads();
}

// ---------------------------------------------------------------------------
// eps = max |(omega/c)^2 - k0^2| : block reduce + atomicMax on float bits
// ---------------------------------------------------------------------------
__global__ void cbs_eps_init(unsigned* e) {
    if (blockIdx.x == 0 && threadIdx.x == 0) *e = 0u;
}

__global__ void cbs_eps_reduce(const float* __restrict__ sos, unsigned* e) {
    __shared__ float red[TPB];
    int tid = threadIdx.x;
    float m = 0.0f;
    for (long i = (long)blockIdx.x * TPB + tid; i < (long)NN * NN;
         i += (long)gridDim.x * TPB) {
        float c = sos[i];
        float v = fabsf(1.0f / (c * c) - K0SQF);
        m = fmaxf(m, v);
    }
    red[tid] = m;
    __syncthreads();
    for (int s = TPB / 2; s > 0; s >>= 1) {
        if (tid < s) red[tid] = fmaxf(red[tid], red[tid + s]);
        __syncthreads();
    }
    if (tid == 0) atomicMax(e, __float_as_uint(red[0]));
}

__global__ void cbs_zero_u(float4* u) {
    long i = (long)blockIdx.x * TPB + threadIdx.x;
    if (i < (long)NN * NN / 2) u[i] = make_float4(0.f, 0.f, 0.f, 0.f);
}

// ---------------------------------------------------------------------------
// Pass 1 (per row): t_row = rowFFT( (vr - i*eps)*u + src )
// u row is staged into LDS by the TDM; result row TDM-stored from LDS.
// ---------------------------------------------------------------------------
__global__ void __launch_bounds__(TPB)
cbs_row_forward(const float* __restrict__ sos, const float* __restrict__ src,
                const float2* __restrict__ u, float2* __restrict__ t,
                const float* __restrict__ epsp) {
    __shared__ float2 A[NN];
    __shared__ float2 B[NN];
    const int  row = blockIdx.x;
    const int  tid = threadIdx.x;
    const float eps = *epsp;
    DftFrag f;
    make_dft16_frags(tid & 31, f);
    const long rb = (long)row * NN;

#if HAVE_TDM
    if (tid < 32) {  // wave 0 drives the tensor DMA (one row, contiguous)
        tdm_tile(u + rb, lds_byte_off(A), NN, 1, NN, 1, NN, false);
        __builtin_amdgcn_s_wait_tensorcnt(0);
    }
    __syncthreads();
#pragma unroll
    for (int r = 0; r < 8; ++r) {
        int   i  = tid + r * 256;
        float c  = sos[rb + i];
        float vr = 1.0f / (c * c) - K0SQF;        // omega = 1
        float  s = src[rb + i];
        float2 uu = A[i];
        A[i] = make_float2(vr * uu.x + eps * uu.y + s, vr * uu.y - eps * uu.x);
    }
#else
#pragma unroll
    for (int r = 0; r < 8; ++r) {
        int   i  = tid + r * 256;
        float c  = sos[rb + i];
        float vr = 1.0f / (c * c) - K0SQF;
        float2 uu = u[rb + i];
        float  s  = src[rb + i];
        A[i] = make_float2(vr * uu.x + eps * uu.y + s, vr * uu.y - eps * uu.x);
    }
#endif
    fft2048<1, 1>(A, B, tid, f);                  // result in B

#if HAVE_TDM
    if (tid < 32) {
        tdm_tile(t + rb, lds_byte_off(B), NN, 1, NN, 1, NN, true);
        __builtin_amdgcn_s_wait_tensorcnt(0);
    }
#else
#pragma unroll
    for (int r = 0; r < 8; ++r) {
        int i = tid + r * 256;
        t[rb + i] = B[i];
    }
#endif
}

// ---------------------------------------------------------------------------
// Pass 2 (per 8-column tile, IN PLACE): t = colIFFT( G .* colFFT(t) )
// TDM gathers the strided 8x2048 tile into LDS (row-major [2048][8]) and
// scatters it back; column FFTs run strided (stride 8) against a contiguous
// ping buffer.  G = 1/(p^2 - k0^2 - i*eps) computed analytically.
// ---------------------------------------------------------------------------
extern __shared__ float2 colsmem[];

__global__ void __launch_bounds__(TPB)
cbs_col_pass(float2* __restrict__ t, const float* __restrict__ epsp) {
    float2* tile = colsmem;            // row-major [NN][8]
    float2* ping = colsmem + 8 * NN;   // contiguous FFT scratch
    const int tid = threadIdx.x;
    const int x0  = blockIdx.x * 8;
    const float eps = *epsp;
    DftFrag f;
    make_dft16_frags(tid & 31, f);

#if HAVE_TDM
    if (tid < 32) {  // 2D tile: 8 elems/row, 2048 rows, row stride NN elems
        tdm_tile(t + x0, lds_byte_off(tile), 8, NN, NN, NN, NN, false);
        __builtin_amdgcn_s_wait_tensorcnt(0);
    }
    __syncthreads();
#else
    for (int l = tid; l < 8 * NN; l += TPB) {
        int c = l & 7;
        int r = l >> 3;
        tile[r * 8 + c] = t[(long)r * NN + x0 + c];
    }
    __syncthreads();
#endif

    for (int c = 0; c < 8; ++c) {
        float2* colA = tile + c;                  // stride-8 column view
        const float px  = kfreq(x0 + c);
        const float px2 = px * px;

        fft2048<8, 1>(colA, ping, tid, f);        // forward FFT -> ping

        // multiply by G, then conjugate (inverse via conj-fft-conj)
#pragma unroll
        for (int r = 0; r < 8; ++r) {
            int   i  = tid + r * 256;
            float py = kfreq(i);
            float dr = px2 + py * py - K0SQF;     // Re(denom); Im(denom) = -eps
            float inv = 1.0f / (dr * dr + eps * eps);
            float gr = dr * inv;                  // G = (dr + i*eps)/|d|^2
            float gi = eps * inv;
            float2 z = ping[i];
            float  zr = z.x * gr - z.y * gi;
            float  zi = z.x * gi + z.y * gr;
            ping[i] = make_float2(zr, -zi);
        }

        fft2048<1, 8>(ping, colA, tid, f);        // fft(conj(.)) -> column

        const float sc = 1.0f / (float)NN;        // conj + 1/N finishes IFFT
#pragma unroll
        for (int r = 0; r < 8; ++r) {
            int i = tid + r * 256;
            float2 z = colA[i * 8];
            colA[i * 8] = make_float2(z.x * sc, -z.y * sc);
        }
    }
    __syncthreads();

#if HAVE_TDM
    if (tid < 32) {
        tdm_tile(t + x0, lds_byte_off(tile), 8, NN, NN, NN, NN, true);
        __builtin_amdgcn_s_wait_tensorcnt(0);
    }
#else
    for (int l = tid; l < 8 * NN; l += TPB) {
        int c = l & 7;
        int r = l >> 3;
        t[(long)r * NN + x0 + c] = tile[r * 8 + c];
    }
#endif
}

// ---------------------------------------------------------------------------
// Pass 3 (per row): Gu = rowIFFT(t_row); u += gamma*(Gu - u), gamma = 1+i*vr/eps
// ---------------------------------------------------------------------------
__global__ void __launch_bounds__(TPB)
cbs_row_inv_update(const float2* __restrict__ t, float2* __restrict__ u,
                   const float* __restrict__ sos, const float* __restrict__ epsp) {
    __shared__ float2 A[NN];
    __shared__ float2 B[NN];
    const int  row = blockIdx.x;
    const int  tid = threadIdx.x;
    const float eps = *epsp;
    const float ie  = 1.0f / eps;
    DftFrag f;
    make_dft16_frags(tid & 31, f);
    const long rb = (long)row * NN;

#if HAVE_TDM
    if (tid < 32) {
        tdm_tile(t + rb, lds_byte_off(A), NN, 1, NN, 1, NN, false);
        __builtin_amdgcn_s_wait_tensorcnt(0);
    }
    __syncthreads();
#pragma unroll
    for (int r = 0; r < 8; ++r) {
        int i = tid + r * 256;
        float2 z = A[i];
        A[i] = make_float2(z.x, -z.y);            // conj for inverse
    }
#else
#pragma unroll
    for (int r = 0; r < 8; ++r) {
        int i = tid + r * 256;
        float2 z = t[rb + i];
        A[i] = make_float2(z.x, -z.y);
    }
#endif
    fft2048<1, 1>(A, B, tid, f);                  // result in B

    const float sc = 1.0f / (float)NN;
#pragma unroll
    for (int r = 0; r < 8; ++r) {
        int   i  = tid + r * 256;
        float c  = sos[rb + i];
        float vr = 1.0f / (c * c) - K0SQF;
        float g  = vr * ie;                       // gamma = 1 + i*g
        float2 z  = B[i];
        float2 Gu = make_float2(z.x * sc, -z.y * sc);
        float2 uu = u[rb + i];
        float dxr = Gu.x - uu.x;
        float dxi = Gu.y - uu.y;
        // u + (1 + i*g)*d
        u[rb + i] = make_float2(uu.x + dxr - g * dxi, uu.y + dxi + g * dxr);
    }
}

__global__ void cbs_copy_out(const float4* __restrict__ u, float4* __restrict__ out) {
    long i = (long)blockIdx.x * TPB + threadIdx.x;
    if (i < (long)NN * NN / 2) out[i] = u[i];     // b128 copies
}

// ---------------------------------------------------------------------------
extern "C" void kernel_launch(void* const* d_in, const int* in_sizes, int n_in,
                              void* d_out, int out_size, void* d_ws, size_t ws_size,
                              hipStream_t stream) {
    const float* sos = (const float*)d_in[0];
    // d_in[1] = pml, unused by the reference forward pass
    const float* src = (const float*)d_in[2];

    const size_t grid_bytes = (size_t)NN * NN * sizeof(float2);
    float2*   u    = (float2*)d_ws;
    float2*   t    = (float2*)((char*)d_ws + grid_bytes);
    unsigned* epsb = (unsigned*)((char*)d_ws + 2 * grid_bytes);
    const float* epsf = (const float*)epsb;

    // eps = max |k^2 - k0^2|
    cbs_eps_init<<<1, 32, 0, stream>>>(epsb);
    cbs_eps_reduce<<<2048, TPB, 0, stream>>>(sos, epsb);

    // u0 = 0
    const long nvec4 = (long)NN * NN / 2;
    cbs_zero_u<<<(nvec4 + TPB - 1) / TPB, TPB, 0, stream>>>((float4*)u);

    const size_t col_lds = (size_t)(8 * NN + NN) * sizeof(float2); // 144 KB < 320 KB WGP LDS
    for (int s = 0; s < NSTG; ++s) {
        cbs_row_forward<<<NN, TPB, 0, stream>>>(sos, src, u, t, epsf);
        cbs_col_pass<<<NN / 8, TPB, col_lds, stream>>>(t, epsf);
        cbs_row_inv_update<<<NN, TPB, 0, stream>>>(t, u, sos, epsf);
    }

    cbs_copy_out<<<(nvec4 + TPB - 1) / TPB, TPB, 0, stream>>>((const float4*)u,
                                                              (float4*)d_out);
}